// TreeLSTM_2233382994142
// MI455X (gfx1250) — compile-verified
//
#include <hip/hip_runtime.h>
#include <math.h>

// ---------------------------------------------------------------------------
// TreeLSTM (event-driven) for gfx1250: fp32 WMMA 16x16x4, wave32, WGP-friendly
// ---------------------------------------------------------------------------

typedef __attribute__((ext_vector_type(2))) float v2f;
typedef __attribute__((ext_vector_type(8))) float v8f;

#define FDIM   128
#define F3     384
#define BSLOT  3
#define NITER  31

// counter array layout (ints): [0..30] nodeCnt, [32..62] edgeCnt,
// [64..94] nodeBase, [96..126] edgeBase, [128..158] nodeFill, [160..190] edgeFill
#define NODE_CNT  0
#define EDGE_CNT  32
#define NODE_BASE 64
#define EDGE_BASE 96
#define NODE_FILL 128
#define EDGE_FILL 160

__device__ __forceinline__ v8f wmma_f32(v2f a, v2f b, v8f c) {
  // D = A(16x4,f32) * B(4x16,f32) + C(16x16,f32)  -> v_wmma_f32_16x16x4_f32
  return __builtin_amdgcn_wmma_f32_16x16x4_f32(false, a, false, b, (short)0, c,
                                               false, false);
}

__device__ __forceinline__ float sigmoidf_(float x) {
  return 1.0f / (1.0f + expf(-x));
}

// ---------------------------------------------------------------------------
__global__ void k_init(float* __restrict__ h, float* __restrict__ c,
                       int* __restrict__ cnts, int total) {
  int stride = gridDim.x * blockDim.x;
  int tid = blockIdx.x * blockDim.x + threadIdx.x;
  for (int i = tid; i < total; i += stride) { h[i] = 0.0f; c[i] = 0.0f; }
  if (tid < 192) cnts[tid] = 0;
}

__global__ void k_hist(const int* __restrict__ node_order,
                       const int* __restrict__ edge_order,
                       const int* __restrict__ adj,
                       int* __restrict__ cnts, int N, int E) {
  int stride = gridDim.x * blockDim.x;
  int tid = blockIdx.x * blockDim.x + threadIdx.x;
  for (int i = tid; i < N; i += stride) {
    int t = node_order[i];
    if (t >= 0 && t < NITER) atomicAdd(&cnts[NODE_CNT + t], 1);
  }
  for (int e = tid; e < E; e += stride) {
    int t = edge_order[e];
    int p = adj[3 * e], ch = adj[3 * e + 1];
    if (t >= 0 && t < NITER && p >= 0 && p < N && ch >= 0 && ch < N)
      atomicAdd(&cnts[EDGE_CNT + t], 1);
  }
}

__global__ void k_scan(int* __restrict__ cnts) {
  if (blockIdx.x == 0 && threadIdx.x == 0) {
    int a = 0;
    for (int t = 0; t < NITER; t++) { cnts[NODE_BASE + t] = a; a += cnts[NODE_CNT + t]; }
    int b = 0;
    for (int t = 0; t < NITER; t++) { cnts[EDGE_BASE + t] = b; b += cnts[EDGE_CNT + t]; }
  }
}

__global__ void k_scatter(const int* __restrict__ node_order,
                          const int* __restrict__ edge_order,
                          const int* __restrict__ adj,
                          int* __restrict__ cnts,
                          int* __restrict__ nlist, int* __restrict__ elist,
                          int N, int E) {
  int stride = gridDim.x * blockDim.x;
  int tid = blockIdx.x * blockDim.x + threadIdx.x;
  for (int i = tid; i < N; i += stride) {
    int t = node_order[i];
    if (t >= 0 && t < NITER) {
      int pos = cnts[NODE_BASE + t] + atomicAdd(&cnts[NODE_FILL + t], 1);
      nlist[pos] = i;
    }
  }
  for (int e = tid; e < E; e += stride) {
    int t = edge_order[e];
    int p = adj[3 * e], ch = adj[3 * e + 1];
    if (t >= 0 && t < NITER && p >= 0 && p < N && ch >= 0 && ch < N) {
      int pos = cnts[EDGE_BASE + t] + atomicAdd(&cnts[EDGE_FILL + t], 1);
      elist[pos] = e;
    }
  }
}

// Zero only the 3 segment slots (2 buffers x 384 floats) of this level's nodes.
__global__ void k_zeroslots(float* __restrict__ flat_h, float* __restrict__ flat_fc,
                            const int* __restrict__ nlist,
                            const int* __restrict__ cnts, int t) {
  const int cnt = cnts[NODE_CNT + t];
  const int base = cnts[NODE_BASE + t];
  int wid = (blockIdx.x * blockDim.x + threadIdx.x) >> 5;
  int lane = threadIdx.x & 31;
  int nw = (gridDim.x * blockDim.x) >> 5;
  for (int i = wid; i < cnt; i += nw) {
    int p = nlist[base + i];
    for (int j = lane; j < F3; j += 32) {
      flat_h[(size_t)p * F3 + j]  = 0.0f;
      flat_fc[(size_t)p * F3 + j] = 0.0f;
    }
  }
}

// ---------------------------------------------------------------------------
// Edge kernel: 16 active edges per group.
//   f  = sigmoid(forest[par]@W_f + b_f + h[child]@U_f)
//   flat_h [par*3+slot] += h[child] ;  flat_fc[par*3+slot] += f * c[child]
// ---------------------------------------------------------------------------
__global__ void __launch_bounds__(128) k_edge(
    const int* __restrict__ adj, const float* __restrict__ forest,
    const float* __restrict__ h, const float* __restrict__ c,
    const float* __restrict__ W_f, const float* __restrict__ b_f,
    const float* __restrict__ U_f,
    float* __restrict__ flat_h, float* __restrict__ flat_fc,
    const int* __restrict__ elist, const int* __restrict__ cnts, int t) {
  __shared__ float Ch[16][132];  // h[child]   (padded: 132 % 64 banks -> conflict-free)
  __shared__ float Cc[16][132];  // c[child]
  __shared__ float Fp[16][132];  // forest[parent]
  __shared__ int ep[16], es[16], ec[16];

  const int cnt  = cnts[EDGE_CNT + t];
  const int base = cnts[EDGE_BASE + t];
  const int tid  = threadIdx.x;
  const int lane = tid & 31;
  const int w    = tid >> 5;            // wave id 0..3
  const int m    = lane & 15;           // A-row / B-col / C-col index
  const int koff = (lane >> 4) << 1;    // K offset half-split (0 or 2)

  for (int g = blockIdx.x; g * 16 < cnt; g += gridDim.x) {
    if (tid < 16) {
      int p = 0, s = 0, chd = 0;
      if (g * 16 + tid < cnt) {
        int e = elist[base + g * 16 + tid];
        p = adj[3 * e]; chd = adj[3 * e + 1];
        int sr = adj[3 * e + 2] + 1;
        s = sr < 0 ? 0 : (sr > 2 ? 2 : sr);   // clip(slot+1, 0, 2)
      }
      ep[tid] = p; es[tid] = s; ec[tid] = chd;
    }
    __syncthreads();
    for (int idx = tid; idx < 16 * FDIM; idx += 128) {
      int r = idx >> 7, j = idx & 127;
      bool act = (g * 16 + r) < cnt;
      int chd = ec[r], p = ep[r];
      Ch[r][j] = act ? h[(size_t)chd * FDIM + j] : 0.0f;
      Cc[r][j] = act ? c[(size_t)chd * FDIM + j] : 0.0f;
      Fp[r][j] = act ? forest[(size_t)p * FDIM + j] : 0.0f;
    }
    __syncthreads();

    // 8 column tiles of 16; 2 per wave.  K = 128 (W_f) + 128 (U_f).
#pragma unroll
    for (int q = 0; q < 2; q++) {
      const int colBase = (w + q * 4) * 16;
      const float bv = b_f[colBase + m];
      v8f acc = {bv, bv, bv, bv, bv, bv, bv, bv};
#pragma unroll 4
      for (int kb = 0; kb < FDIM; kb += 4) {
        v2f a, bm;
        a.x  = Fp[m][kb + koff];
        a.y  = Fp[m][kb + koff + 1];
        bm.x = W_f[(kb + koff) * FDIM + colBase + m];
        bm.y = W_f[(kb + koff + 1) * FDIM + colBase + m];
        acc = wmma_f32(a, bm, acc);
      }
#pragma unroll 4
      for (int kb = 0; kb < FDIM; kb += 4) {
        v2f a, bm;
        a.x  = Ch[m][kb + koff];
        a.y  = Ch[m][kb + koff + 1];
        bm.x = U_f[(kb + koff) * FDIM + colBase + m];
        bm.y = U_f[(kb + koff + 1) * FDIM + colBase + m];
        acc = wmma_f32(a, bm, acc);
      }
      // f = sigmoid(acc); scatter f*child_c into segment buffer
#pragma unroll
      for (int v = 0; v < 8; v++) {
        int row = v + ((lane >> 4) << 3);
        int col = colBase + m;
        if (g * 16 + row < cnt) {
          float fg = sigmoidf_(acc[v]);
          float fc = fg * Cc[row][col];
          atomicAdd(&flat_fc[((size_t)ep[row] * BSLOT + es[row]) * FDIM + col], fc);
        }
      }
    }
    // scatter child_h into segment buffer
    for (int idx = tid; idx < 16 * FDIM; idx += 128) {
      int r = idx >> 7, j = idx & 127;
      if (g * 16 + r < cnt)
        atomicAdd(&flat_h[((size_t)ep[r] * BSLOT + es[r]) * FDIM + j], Ch[r][j]);
    }
    __syncthreads();
  }
}

// ---------------------------------------------------------------------------
// Node kernel: 16 active nodes per group.
//   iou   = forest@W_iou + merged_h@U_iou + b_iou      (K = 128 + 384, N = 384)
//   c_red = merged_fc@W_c + b_c                        (K = 384, N = 128)
//   new_c = sig(i)*tanh(u) + c_red ; new_h = sig(o)*tanh(new_c)
// ---------------------------------------------------------------------------
__global__ void __launch_bounds__(128) k_node(
    const float* __restrict__ forest, float* __restrict__ h, float* __restrict__ c,
    const float* __restrict__ W_iou, const float* __restrict__ b_iou,
    const float* __restrict__ U_iou,
    const float* __restrict__ W_c, const float* __restrict__ b_c,
    const float* __restrict__ flat_h, const float* __restrict__ flat_fc,
    const int* __restrict__ nlist, const int* __restrict__ cnts, int t) {
  __shared__ float Xh[16][388];   // merged child-h; reused for activations i|o|u
  __shared__ float Xfc[16][388];  // merged f*child_c
  __shared__ float Xg[16][132];   // forest row; reused for c_reduce
  __shared__ int np[16];

  const int cnt  = cnts[NODE_CNT + t];
  const int base = cnts[NODE_BASE + t];
  const int tid  = threadIdx.x;
  const int lane = tid & 31;
  const int w    = tid >> 5;
  const int m    = lane & 15;
  const int koff = (lane >> 4) << 1;

  for (int g = blockIdx.x; g * 16 < cnt; g += gridDim.x) {
    if (tid < 16) np[tid] = (g * 16 + tid < cnt) ? nlist[base + g * 16 + tid] : 0;
    __syncthreads();
    for (int idx = tid; idx < 16 * F3; idx += 128) {
      int r = idx / F3, j = idx % F3;
      bool act = (g * 16 + r) < cnt;
      int p = np[r];
      Xh[r][j]  = act ? flat_h[(size_t)p * F3 + j] : 0.0f;
      Xfc[r][j] = act ? flat_fc[(size_t)p * F3 + j] : 0.0f;
    }
    for (int idx = tid; idx < 16 * FDIM; idx += 128) {
      int r = idx >> 7, j = idx & 127;
      bool act = (g * 16 + r) < cnt;
      Xg[r][j] = act ? forest[(size_t)np[r] * FDIM + j] : 0.0f;
    }
    __syncthreads();

    // GEMM1: 24 column tiles over 4 waves -> 6 per wave, accumulators in VGPRs
    v8f accs[6];
#pragma unroll
    for (int q = 0; q < 6; q++) {
      const int colBase = (q * 4 + w) * 16;
      const float bv = b_iou[colBase + m];
      v8f acc = {bv, bv, bv, bv, bv, bv, bv, bv};
#pragma unroll 4
      for (int kb = 0; kb < FDIM; kb += 4) {           // forest @ W_iou
        v2f a, bm;
        a.x  = Xg[m][kb + koff];
        a.y  = Xg[m][kb + koff + 1];
        bm.x = W_iou[(kb + koff) * F3 + colBase + m];
        bm.y = W_iou[(kb + koff + 1) * F3 + colBase + m];
        acc = wmma_f32(a, bm, acc);
      }
#pragma unroll 4
      for (int kb = 0; kb < F3; kb += 4) {             // merged_h @ U_iou
        v2f a, bm;
        a.x  = Xh[m][kb + koff];
        a.y  = Xh[m][kb + koff + 1];
        bm.x = U_iou[(kb + koff) * F3 + colBase + m];
        bm.y = U_iou[(kb + koff + 1) * F3 + colBase + m];
        acc = wmma_f32(a, bm, acc);
      }
      accs[q] = acc;
    }
    __syncthreads();  // everyone done reading Xh/Xg -> safe to reuse

    // activations written back into Xh: cols 0-255 sigmoid (i,o), 256-383 tanh (u)
#pragma unroll
    for (int q = 0; q < 6; q++) {
      const int colBase = (q * 4 + w) * 16;
#pragma unroll
      for (int v = 0; v < 8; v++) {
        int row = v + ((lane >> 4) << 3);
        int col = colBase + m;
        float x = accs[q][v];
        Xh[row][col] = (col < 256) ? sigmoidf_(x) : tanhf(x);
      }
    }
    __syncthreads();

    // GEMM2: c_reduce = Xfc @ W_c + b_c -> Xg   (8 tiles, 2 per wave)
#pragma unroll
    for (int q = 0; q < 2; q++) {
      const int colBase = (q * 4 + w) * 16;
      const float bv = b_c[colBase + m];
      v8f acc = {bv, bv, bv, bv, bv, bv, bv, bv};
#pragma unroll 4
      for (int kb = 0; kb < F3; kb += 4) {
        v2f a, bm;
        a.x  = Xfc[m][kb + koff];
        a.y  = Xfc[m][kb + koff + 1];
        bm.x = W_c[(kb + koff) * FDIM + colBase + m];
        bm.y = W_c[(kb + koff + 1) * FDIM + colBase + m];
        acc = wmma_f32(a, bm, acc);
      }
#pragma unroll
      for (int v = 0; v < 8; v++) {
        int row = v + ((lane >> 4) << 3);
        Xg[row][colBase + m] = acc[v];
      }
    }
    __syncthreads();

    // elementwise LSTM update + writeout
    for (int r = 0; r < 16; r++) {
      if (g * 16 + r < cnt) {
        int p = np[r];
        int j = tid;  // 128 threads = 128 features
        float iv = Xh[r][j], ov = Xh[r][128 + j], uv = Xh[r][256 + j];
        float nc = iv * uv + Xg[r][j];
        float nh = ov * tanhf(nc);
        c[(size_t)p * FDIM + j] = nc;
        h[(size_t)p * FDIM + j] = nh;
      }
    }
    __syncthreads();
  }
}

// ---------------------------------------------------------------------------
extern "C" void kernel_launch(void* const* d_in, const int* in_sizes, int n_in,
                              void* d_out, int out_size, void* d_ws, size_t ws_size,
                              hipStream_t stream) {
  const float* forest     = (const float*)d_in[0];
  const int*   adj        = (const int*)d_in[1];
  const int*   node_order = (const int*)d_in[2];
  const int*   edge_order = (const int*)d_in[3];
  const float* W_iou      = (const float*)d_in[4];
  const float* b_iou      = (const float*)d_in[5];
  const float* U_iou      = (const float*)d_in[6];
  const float* W_c        = (const float*)d_in[7];
  const float* b_c        = (const float*)d_in[8];
  const float* W_f        = (const float*)d_in[9];
  const float* b_f        = (const float*)d_in[10];
  const float* U_f        = (const float*)d_in[11];
  const int N = in_sizes[2];
  const int E = in_sizes[3];
  float* h = (float*)d_out;
  (void)n_in; (void)out_size; (void)ws_size;

  char* ws = (char*)d_ws;
  size_t off = 0;
  auto alloc = [&](size_t bytes) -> void* {
    void* p = ws + off;
    off = (off + bytes + 255) & ~(size_t)255;
    return p;
  };
  float* c       = (float*)alloc((size_t)N * FDIM * sizeof(float));
  float* flat_h  = (float*)alloc((size_t)N * F3 * sizeof(float));
  float* flat_fc = (float*)alloc((size_t)N * F3 * sizeof(float));
  int*   nlist   = (int*)alloc((size_t)N * sizeof(int));
  int*   elist   = (int*)alloc((size_t)E * sizeof(int));
  int*   cnts    = (int*)alloc(192 * sizeof(int));

  k_init<<<dim3(2048), dim3(256), 0, stream>>>(h, c, cnts, N * FDIM);
  k_hist<<<dim3(1024), dim3(256), 0, stream>>>(node_order, edge_order, adj, cnts, N, E);
  k_scan<<<dim3(1), dim3(1), 0, stream>>>(cnts);
  k_scatter<<<dim3(1024), dim3(256), 0, stream>>>(node_order, edge_order, adj, cnts,
                                                  nlist, elist, N, E);
  for (int t = 0; t < NITER; t++) {
    k_zeroslots<<<dim3(512), dim3(256), 0, stream>>>(flat_h, flat_fc, nlist, cnts, t);
    k_edge<<<dim3(512), dim3(128), 0, stream>>>(adj, forest, h, c, W_f, b_f, U_f,
                                                flat_h, flat_fc, elist, cnts, t);
    k_node<<<dim3(512), dim3(128), 0, stream>>>(forest, h, c, W_iou, b_iou, U_iou,
                                                W_c, b_c, flat_h, flat_fc,
                                                nlist, cnts, t);
  }
}